// GCN_40054865002827
// MI455X (gfx1250) — compile-verified
//
#include <hip/hip_runtime.h>
#include <hip/hip_bf16.h>

typedef __attribute__((ext_vector_type(2))) float v2f;
typedef __attribute__((ext_vector_type(8))) float v8f;
typedef int v4i __attribute__((vector_size(4 * sizeof(int))));

#define AS_GLOBAL __attribute__((address_space(1)))
#define AS_LDS    __attribute__((address_space(3)))

#if defined(__has_builtin)
#if __has_builtin(__builtin_amdgcn_global_load_async_to_lds_b128)
#define HAVE_ASYNC_LDS_COPY 1
#endif
#endif

// ---------------------------------------------------------------------------
// Degree / symmetric normalization:  dinv[i] = (deg_in[i] + 1)^-1/2
// ---------------------------------------------------------------------------
__global__ void k_zero(float* __restrict__ p, int n) {
  int i = blockIdx.x * blockDim.x + threadIdx.x;
  if (i < n) p[i] = 0.0f;
}

__global__ void k_count_deg(const int* __restrict__ dst, float* __restrict__ deg, int E) {
  int e = blockIdx.x * blockDim.x + threadIdx.x;
  if (e < E) atomicAdd(&deg[dst[e]], 1.0f);
}

__global__ void k_dinv(float* __restrict__ deg, int N) {
  int i = blockIdx.x * blockDim.x + threadIdx.x;
  if (i < N) deg[i] = rsqrtf(deg[i] + 1.0f);
}

// ---------------------------------------------------------------------------
// Pack W[K,16] into WMMA-B-native order so each lane's B operand is one
// contiguous float2:
//   Bp[g*64 + lane*2 + s] = W[(4g + 2*(lane>>4) + s)*16 + (lane&15)], s=0,1
// (B 4x16 f32 layout: VGPR0 = K rows {0|2}, VGPR1 = K rows {1|3}, N=lane&15)
// ---------------------------------------------------------------------------
__global__ void k_pack_w(const float* __restrict__ W, float* __restrict__ Bp,
                         int K /* multiple of 4 */) {
  int t = blockIdx.x * blockDim.x + threadIdx.x;   // one thread per (g,lane)
  int total = (K / 4) * 32;
  if (t >= total) return;
  int g = t >> 5;
  int l = t & 31;
  int khalf = (l >> 4) * 2;
  int sub = l & 15;
  Bp[2 * t + 0] = W[(size_t)(4 * g + khalf + 0) * 16 + sub];
  Bp[2 * t + 1] = W[(size_t)(4 * g + khalf + 1) * 16 + sub];
}

// ---------------------------------------------------------------------------
// Layer-1 GEMM: h[N,16] = x[N,K] @ W[K,16] via V_WMMA_F32_16X16X4_F32.
//  - Packed W (K*16 floats = 256 KB for K=4000) is staged once per WG into
//    LDS (async-to-LDS if available), then read conflict-free via ds_load_b64:
//    lane l touches LDS words {2l, 2l+1} -> all 64 banks exactly once.
//  - x is streamed non-temporally (800 MB, read exactly once -> HBM-bound,
//    AI = 8 FLOP/B, floor ~34 us @ 23.3 TB/s). A operand: one b64/lane/step.
//  - 512 threads = 16 wave32 per WG, one 16x16 output tile per wave.
//  - Tile-bound exit is wave-uniform and after the barrier, so EXEC==all-ones
//    at every WMMA (ISA requirement).
// ---------------------------------------------------------------------------
__global__ void __launch_bounds__(512)
k_gemm_f32_wmma(const float* __restrict__ x,
                const float* __restrict__ Bp,   // packed W, [K/4][32][2] floats
                float* __restrict__ h,          // [N,16]
                int N, int K) {
  extern __shared__ float ldsW[];                // K*16 floats
  const int tid = threadIdx.x;
  const int nv4 = (K * 16) >> 2;                 // float4 count

#if HAVE_ASYNC_LDS_COPY
  for (int i = tid; i < nv4; i += blockDim.x) {
    __builtin_amdgcn_global_load_async_to_lds_b128(
        (AS_GLOBAL v4i*)(Bp + (size_t)i * 4),
        (AS_LDS v4i*)(ldsW + (size_t)i * 4),
        0, 0);
  }
  asm volatile("s_wait_asynccnt 0" ::: "memory");
#else
  {
    const float4* srcv = (const float4*)Bp;
    float4* dstv = (float4*)ldsW;
    for (int i = tid; i < nv4; i += blockDim.x) dstv[i] = srcv[i];
  }
#endif
  __syncthreads();

  const int lane = tid & 31;
  const int wave = tid >> 5;
  const int tile = blockIdx.x * (blockDim.x >> 5) + wave;
  const int row0 = tile * 16;
  if (row0 >= N) return;                         // uniform across the wave

  const int sub   = lane & 15;                   // A row within tile
  const int khalf = (lane >> 4) * 2;             // A K sub-offset {0,2}
  const float* xp = x + (size_t)(row0 + sub) * (size_t)K + khalf;
  const float* bp = ldsW + lane * 2;

  v8f acc = {};
  const int G = K >> 2;
#pragma unroll 8
  for (int g = 0; g < G; ++g) {
    v2f a = __builtin_nontemporal_load((const v2f*)(xp + 4 * g)); // x[row][4g+khalf..+1]
    v2f b = *(const v2f*)(bp + 64 * g);                           // ds_load_b64
    acc = __builtin_amdgcn_wmma_f32_16x16x4_f32(
        /*neg_a=*/false, a, /*neg_b=*/false, b,
        /*c_mod=*/(short)0, acc, /*reuse_a=*/false, /*reuse_b=*/false);
  }

  // C/D layout: vgpr r = row (r + 8*(lane>>4)), col = lane&15
  const int mb = (lane >> 4) * 8;
#pragma unroll
  for (int r = 0; r < 8; ++r)
    h[(size_t)(row0 + mb + r) * 16 + sub] = acc[r];
}

// ---------------------------------------------------------------------------
// agg[i,f] = dinv[i]^2 * h[i,f]   (self-loop message, also initializes agg)
// ---------------------------------------------------------------------------
template <int F>
__global__ void k_self_scale(const float* __restrict__ h, const float* __restrict__ dinv,
                             float* __restrict__ agg, int N) {
  int t = blockIdx.x * blockDim.x + threadIdx.x;
  if (t < N * F) {
    int i = t / F;
    float d = dinv[i];
    agg[t] = d * d * h[t];
  }
}

// ---------------------------------------------------------------------------
// Edge scatter: agg[dst,f] += dinv[src]*dinv[dst] * h[src,f]
// h/agg feature maps are L2-resident (3.2 MB), atomics are non-returning.
// ---------------------------------------------------------------------------
template <int F>
__global__ void k_edge_agg(const int* __restrict__ src, const int* __restrict__ dst,
                           const float* __restrict__ dinv, const float* __restrict__ h,
                           float* __restrict__ agg, int E) {
  int t = blockIdx.x * blockDim.x + threadIdx.x;
  if (t >= E * F) return;
  int e = t / F;
  int f = t % F;
  int s = src[e], d = dst[e];
  float norm = dinv[s] * dinv[d];
  atomicAdd(&agg[(size_t)d * F + f], norm * h[(size_t)s * F + f]);
}

// ---------------------------------------------------------------------------
// In-place bias + ReLU over [N,F]
// ---------------------------------------------------------------------------
template <int F>
__global__ void k_bias_relu(float* __restrict__ a, const float* __restrict__ b, int N) {
  int t = blockIdx.x * blockDim.x + threadIdx.x;
  if (t < N * F) {
    float v = a[t] + b[t % F];
    a[t] = v > 0.0f ? v : 0.0f;
  }
}

// ---------------------------------------------------------------------------
// Layer-2 dense: h2[i,0:2] = z[i,0:16] @ W2[16,2]   (tiny, VALU is fine)
// ---------------------------------------------------------------------------
__global__ void k_fc16x2(const float* __restrict__ z, const float* __restrict__ W2,
                         float* __restrict__ h2, int N) {
  int i = blockIdx.x * blockDim.x + threadIdx.x;
  if (i >= N) return;
  const float* zi = z + (size_t)i * 16;
  float a0 = 0.0f, a1 = 0.0f;
#pragma unroll
  for (int f = 0; f < 16; ++f) {
    float v = zi[f];
    a0 += v * W2[f * 2 + 0];
    a1 += v * W2[f * 2 + 1];
  }
  h2[(size_t)i * 2 + 0] = a0;
  h2[(size_t)i * 2 + 1] = a1;
}

// ---------------------------------------------------------------------------
// out[i,:] = log_softmax(agg2[i,:] + b2)  with C==2 closed form
// ---------------------------------------------------------------------------
__global__ void k_logsoftmax2(const float* __restrict__ agg, const float* __restrict__ b2,
                              float* __restrict__ out, int N) {
  int i = blockIdx.x * blockDim.x + threadIdx.x;
  if (i >= N) return;
  float a0 = agg[(size_t)i * 2 + 0] + b2[0];
  float a1 = agg[(size_t)i * 2 + 1] + b2[1];
  float m = fmaxf(a0, a1);
  float lse = m + logf(expf(a0 - m) + expf(a1 - m));
  out[(size_t)i * 2 + 0] = a0 - lse;
  out[(size_t)i * 2 + 1] = a1 - lse;
}

// ---------------------------------------------------------------------------
extern "C" void kernel_launch(void* const* d_in, const int* in_sizes, int n_in,
                              void* d_out, int out_size, void* d_ws, size_t ws_size,
                              hipStream_t stream) {
  const float* x  = (const float*)d_in[0];
  const int*   ei = (const int*)d_in[1];      // edge_index [2,E] flat: src then dst
  const float* W1 = (const float*)d_in[2];
  const float* b1 = (const float*)d_in[3];
  const float* W2 = (const float*)d_in[4];
  const float* b2 = (const float*)d_in[5];

  const int H   = in_sizes[3];                // 16
  const int Fin = in_sizes[2] / H;            // 4000
  const int N   = in_sizes[0] / Fin;          // 50000
  const int E   = in_sizes[1] / 2;            // 1.6M
  const int* src = ei;
  const int* dst = ei + E;

  // Workspace layout (floats): h1 | agg1 | h2 | agg2 | dinv | packed W1
  float* ws   = (float*)d_ws;
  float* h1   = ws;  ws += (size_t)N * H;
  float* agg1 = ws;  ws += (size_t)N * H;
  float* h2   = ws;  ws += (size_t)N * 2;
  float* agg2 = ws;  ws += (size_t)N * 2;
  float* dinv = ws;  ws += N;
  float* Bp   = ws;  ws += (size_t)Fin * H;   // WMMA-packed W1

  const int B = 256;
  auto cdiv = [](long long a, long long b) { return (int)((a + b - 1) / b); };

  // normalization coefficients
  k_zero<<<cdiv(N, B), B, 0, stream>>>(dinv, N);
  k_count_deg<<<cdiv(E, B), B, 0, stream>>>(dst, dinv, E);
  k_dinv<<<cdiv(N, B), B, 0, stream>>>(dinv, N);

  // layer 1: pack W, then WMMA GEMM (HBM-bound: streams 800MB of x once)
  k_pack_w<<<cdiv((long long)(Fin / 4) * 32, B), B, 0, stream>>>(W1, Bp, Fin);
  const int tiles = (N + 15) / 16;
  const int thrPerBlk = 512;                  // 16 wave32 -> 16 tiles per WG
  const int tilesPerBlk = thrPerBlk / 32;
  const size_t ldsBytes = (size_t)Fin * H * sizeof(float);   // 256 KB <= 320 KB
  k_gemm_f32_wmma<<<cdiv(tiles, tilesPerBlk), thrPerBlk, ldsBytes, stream>>>(x, Bp, h1, N, Fin);

  k_self_scale<16><<<cdiv((long long)N * 16, B), B, 0, stream>>>(h1, dinv, agg1, N);
  k_edge_agg<16><<<cdiv((long long)E * 16, B), B, 0, stream>>>(src, dst, dinv, h1, agg1, E);
  k_bias_relu<16><<<cdiv((long long)N * 16, B), B, 0, stream>>>(agg1, b1, N);

  // layer 2 (tiny)
  k_fc16x2<<<cdiv(N, B), B, 0, stream>>>(agg1, W2, h2, N);
  k_self_scale<2><<<cdiv((long long)N * 2, B), B, 0, stream>>>(h2, dinv, agg2, N);
  k_edge_agg<2><<<cdiv((long long)E * 2, B), B, 0, stream>>>(src, dst, dinv, h2, agg2, E);
  k_logsoftmax2<<<cdiv(N, B), B, 0, stream>>>(agg2, b2, (float*)d_out, N);
}